// RGCN_43817256354376
// MI455X (gfx1250) — compile-verified
//
#include <hip/hip_runtime.h>

// ---------------------------------------------------------------------------
// RGCN on MI455X (gfx1250), fp32 via V_WMMA_F32_16X16X4_F32.
//
// Trick: mean-aggregation commutes with the relation weight (message is
// linear in x[src]), so per conv we do ONE edge scatter pass of raw features
// into agg[N, R*HID], then ONE K=512 WMMA GEMM with the R weight matrices
// stacked along K (root term fused as a second K=128 phase).
//
// B tile is stored in LDS in a WMMA-operand-swizzled layout so each lane's
// (b.x, b.y) register pair is one aligned ds_load_b64 (no shuffle movs),
// with +32-float group padding so the two half-waves use disjoint banks.
// All NT B operands of a ks-step are loaded into registers before the WMMA
// burst so several ds_loads stay in flight (fewer s_wait_dscnt drains).
// ---------------------------------------------------------------------------

#define NA_    100000
#define NB_    100000
#define NTOT_  200000
#define E_     600000
#define HID_   128
#define OUTD_  64
#define RREL_  4

typedef __attribute__((ext_vector_type(2))) float v2f;
typedef __attribute__((ext_vector_type(8))) float v8f;

// ---------------------------------------------------------------------------
// zero fill (float4 vectorized)
// ---------------------------------------------------------------------------
__global__ void zero_f4(float4* __restrict__ p, int n4) {
  int i = blockIdx.x * 256 + threadIdx.x;
  if (i < n4) p[i] = make_float4(0.f, 0.f, 0.f, 0.f);
}

// ---------------------------------------------------------------------------
// W[r] = sum_b comp[r,b] * basis[b]   (stacked along K: Wcat[r*HID+i, o])
// ---------------------------------------------------------------------------
__global__ void combine_w(const float* __restrict__ comp,
                          const float* __restrict__ basis,
                          float* __restrict__ W, int per, int total) {
  int i = blockIdx.x * 256 + threadIdx.x;
  if (i >= total) return;
  int r = i / per;
  int rem = i - r * per;
  float acc = 0.f;
#pragma unroll
  for (int b = 0; b < 4; ++b)
    acc += comp[r * 4 + b] * basis[(size_t)b * per + rem];
  W[i] = acc;
}

// ---------------------------------------------------------------------------
// edge scatter: agg[dst, r*128 + c] += x[src, c];  cnt[dst, r] += 1
// one wave per edge, one float4 per lane (128 floats/row)
// ---------------------------------------------------------------------------
__global__ __launch_bounds__(256) void edge_agg(
    const float* __restrict__ x, const int* __restrict__ src,
    const int* __restrict__ dst, const int* __restrict__ et,
    float* __restrict__ agg, float* __restrict__ cnt, int E) {
  int lane = threadIdx.x & 31;
  int e = blockIdx.x * 8 + (threadIdx.x >> 5);
  if (e >= E) return;
  int s = src[e];
  int d = dst[e];
  int r = et[e];
  float4 v = ((const float4*)(x + (size_t)s * HID_))[lane];
  float* ab = agg + (size_t)d * (RREL_ * HID_) + r * HID_ + lane * 4;
  atomicAdd(ab + 0, v.x);
  atomicAdd(ab + 1, v.y);
  atomicAdd(ab + 2, v.z);
  atomicAdd(ab + 3, v.w);
  if (lane == 0) atomicAdd(cnt + (size_t)d * RREL_ + r, 1.0f);
}

// ---------------------------------------------------------------------------
// Tiled fp32 WMMA GEMM:
//   out[M, NC] = relu?( phase0: scale(A[M,K]) @ B[K,NC]
//                     + phase1: A2[M,K2]      @ B2[K2,NC]  + bias )
// phase0 scale: A[v, k] *= 1/max(cnt[v, k>>7], 1)   (per-relation mean)
// Block: 256 threads = 8 waves; 128 rows/block; wave w owns rows w*16..w*16+15
// and all NT column tiles.
//
// LDS:
//   As: 128 rows x 64 k, row stride 66 (A operand = one b64 per lane)
//   Bs: swizzled. K-chunk row r = 4g + 2h + p (g=k-group, h=half-wave,
//       p=parity) stored at (2g+h)*BSTR + col*2 + p, BSTR = NC*2 + 32.
//       -> B operand (rows k+2h, k+2h+1 at one col) = one aligned b64 load,
//          half-waves on disjoint bank halves.
// ---------------------------------------------------------------------------
template <int NT>
__global__ __launch_bounds__(256) void gemm_wmma_k(
    const float* __restrict__ A, int lda, int K,
    const float* __restrict__ cntp,                       // [M,4] or null
    const float* __restrict__ A2, int lda2, int K2,       // root input or null
    const float* __restrict__ B,                          // [K, NC]
    const float* __restrict__ B2,                         // [K2, NC]
    const float* __restrict__ bias,                       // [NC] or null
    float* __restrict__ out, int ldo, int M, int relu) {
  constexpr int NC = NT * 16;
  constexpr int BSTR = NC * 2 + 32;   // floats per (group,half) B row-pair
  __shared__ float As[128 * 66];
  __shared__ float Bs[32 * BSTR];

  const int tid = threadIdx.x;
  const int lane = tid & 31;
  const int w = tid >> 5;
  const int hi = lane >> 4;   // half-wave select
  const int lo = lane & 15;
  const int mrow0 = blockIdx.x * 128;

  v8f acc[NT];
#pragma unroll
  for (int t = 0; t < NT; ++t)
#pragma unroll
    for (int j = 0; j < 8; ++j) acc[t][j] = 0.0f;

  for (int phase = 0; phase < 2; ++phase) {
    const float* Ap = phase ? A2 : A;
    const float* Bp = phase ? B2 : B;
    const int Kp = phase ? K2 : K;
    const int ldap = phase ? lda2 : lda;
    const bool scaled = (phase == 0) && (cntp != nullptr);
    if (Ap == nullptr || Kp <= 0) continue;

    for (int kbase = 0; kbase < Kp; kbase += 64) {
      // ---- stage A chunk: 128 rows x 64 k (coalesced along k) ----
      for (int i = tid; i < 128 * 64; i += 256) {
        int row = i >> 6;
        int kk = i & 63;
        int gr = mrow0 + row;
        int gk = kbase + kk;
        float v = 0.0f;
        if (gr < M && gk < Kp) {
          v = Ap[(size_t)gr * ldap + gk];
          if (scaled) {
            float c = cntp[(size_t)gr * 4 + (gk >> 7)];
            v *= (1.0f / fmaxf(c, 1.0f));
          }
        }
        As[row * 66 + kk] = v;
      }
      // ---- stage B chunk: 64 k x NC cols, swizzled for b64 operand loads --
      for (int i = tid; i < 64 * NC; i += 256) {
        int row = i / NC;              // k within chunk
        int col = i - row * NC;
        int gk = kbase + row;
        float v = (gk < Kp) ? Bp[(size_t)gk * NC + col] : 0.0f;
        int g = row >> 2;
        int h = (row >> 1) & 1;
        int p = row & 1;
        Bs[(g * 2 + h) * BSTR + col * 2 + p] = v;
      }
      __syncthreads();

      // ---- 16 WMMA K-steps of 4 per chunk ----
      const int arow = (w * 16 + lo) * 66;
#pragma unroll
      for (int ks = 0; ks < 64; ks += 4) {
        // A 16x4 f32 layout: lanes 0-15 K={ks,ks+1}, lanes 16-31 K={ks+2,ks+3}
        v2f a = *(const v2f*)&As[arow + ks + 2 * hi];
        const int bbase = ((ks >> 1) + hi) * BSTR + lo * 2;
        // batch all B operand loads so several ds_loads stay in flight
        v2f b[NT];
#pragma unroll
        for (int nt = 0; nt < NT; ++nt)
          b[nt] = *(const v2f*)&Bs[bbase + nt * 32];
#pragma unroll
        for (int nt = 0; nt < NT; ++nt)
          acc[nt] = __builtin_amdgcn_wmma_f32_16x16x4_f32(
              false, a, false, b[nt], (short)0, acc[nt], false, false);
      }
      __syncthreads();
    }
  }

  // ---- epilogue: C/D layout lane lo = col, VGPR j -> row j + 8*hi ----
#pragma unroll
  for (int nt = 0; nt < NT; ++nt) {
    int col = nt * 16 + lo;
    float bv = bias ? bias[col] : 0.0f;
#pragma unroll
    for (int j = 0; j < 8; ++j) {
      int row = mrow0 + w * 16 + hi * 8 + j;
      if (row < M) {
        float v = acc[nt][j] + bv;
        if (relu) v = fmaxf(v, 0.0f);
        out[(size_t)row * ldo + col] = v;
      }
    }
  }
}

// ---------------------------------------------------------------------------
extern "C" void kernel_launch(void* const* d_in, const int* in_sizes, int n_in,
                              void* d_out, int out_size, void* d_ws,
                              size_t ws_size, hipStream_t stream) {
  (void)in_sizes; (void)n_in; (void)out_size; (void)ws_size;

  const float* x_author = (const float*)d_in[0];
  const float* x_paper  = (const float*)d_in[1];
  const int*   src      = (const int*)d_in[2];
  const int*   dst      = (const int*)d_in[3];
  const int*   et       = (const int*)d_in[4];
  const float* Wpa      = (const float*)d_in[5];
  const float* bpa      = (const float*)d_in[6];
  const float* Wpb      = (const float*)d_in[7];
  const float* bpb      = (const float*)d_in[8];
  const float* basis0   = (const float*)d_in[9];
  const float* comp0    = (const float*)d_in[10];
  const float* basis1   = (const float*)d_in[11];
  const float* comp1    = (const float*)d_in[12];
  const float* root1    = (const float*)d_in[13];
  const float* basis2   = (const float*)d_in[14];
  const float* comp2    = (const float*)d_in[15];
  const float* root2    = (const float*)d_in[16];

  // d_out tuple layout: (x[N,64], lat0[N,128], lat1[N,128], lat2[N,128])
  float* out  = (float*)d_out;
  float* xfin = out;
  float* lat0 = xfin + (size_t)NTOT_ * OUTD_;
  float* lat1 = lat0 + (size_t)NTOT_ * HID_;
  float* lat2 = lat1 + (size_t)NTOT_ * HID_;

  // workspace: agg[N,512] | cnt[N,4] | Wcat0[512,128] | Wcat1[512,128] | Wcat2[512,64]
  float* ws  = (float*)d_ws;
  float* agg = ws;
  float* cnt = agg + (size_t)NTOT_ * (RREL_ * HID_);
  float* W0  = cnt + (size_t)NTOT_ * RREL_;
  float* W1  = W0 + (size_t)RREL_ * HID_ * HID_;
  float* W2  = W1 + (size_t)RREL_ * HID_ * HID_;

  // --- combine basis weights (tiny) ---
  {
    int per = HID_ * HID_, tot = RREL_ * per;
    combine_w<<<(tot + 255) / 256, 256, 0, stream>>>(comp0, basis0, W0, per, tot);
    combine_w<<<(tot + 255) / 256, 256, 0, stream>>>(comp1, basis1, W1, per, tot);
    int per2 = HID_ * OUTD_, tot2 = RREL_ * per2;
    combine_w<<<(tot2 + 255) / 256, 256, 0, stream>>>(comp2, basis2, W2, per2, tot2);
  }

  const int gbA = (NA_ + 127) / 128;    // 782
  const int gbN = (NTOT_ + 127) / 128;  // 1563
  const int gbE = (E_ + 7) / 8;         // 75000
  const int zr4 = (int)(((size_t)NTOT_ * (RREL_ * HID_ + RREL_)) / 4);
  const int gbZ = (zr4 + 255) / 256;

  // --- per-type projections + relu -> lat0 ---
  gemm_wmma_k<8><<<gbA, 256, 0, stream>>>(
      x_author, 64, 64, nullptr, nullptr, 0, 0, Wpa, nullptr, bpa,
      lat0, HID_, NA_, 1);
  gemm_wmma_k<8><<<gbA, 256, 0, stream>>>(
      x_paper, 96, 96, nullptr, nullptr, 0, 0, Wpb, nullptr, bpb,
      lat0 + (size_t)NA_ * HID_, HID_, NB_, 1);

  // --- conv0 (no root) -> lat1 ---
  zero_f4<<<gbZ, 256, 0, stream>>>((float4*)agg, zr4);
  edge_agg<<<gbE, 256, 0, stream>>>(lat0, src, dst, et, agg, cnt, E_);
  gemm_wmma_k<8><<<gbN, 256, 0, stream>>>(
      agg, RREL_ * HID_, RREL_ * HID_, cnt, nullptr, 0, 0, W0, nullptr,
      nullptr, lat1, HID_, NTOT_, 1);

  // --- conv1 (+root1) -> lat2 ---
  zero_f4<<<gbZ, 256, 0, stream>>>((float4*)agg, zr4);
  edge_agg<<<gbE, 256, 0, stream>>>(lat1, src, dst, et, agg, cnt, E_);
  gemm_wmma_k<8><<<gbN, 256, 0, stream>>>(
      agg, RREL_ * HID_, RREL_ * HID_, cnt, lat1, HID_, HID_, W1, root1,
      nullptr, lat2, HID_, NTOT_, 1);

  // --- conv2 (+root2, no relu, OUT=64) -> xfin ---
  zero_f4<<<gbZ, 256, 0, stream>>>((float4*)agg, zr4);
  edge_agg<<<gbE, 256, 0, stream>>>(lat2, src, dst, et, agg, cnt, E_);
  gemm_wmma_k<4><<<gbN, 256, 0, stream>>>(
      agg, RREL_ * HID_, RREL_ * HID_, cnt, lat2, HID_, HID_, W2, root2,
      nullptr, xfin, OUTD_, NTOT_, 0);
}